// VarianceAdaptor_52742198395189
// MI455X (gfx1250) — compile-verified
//
#include <hip/hip_runtime.h>

typedef __attribute__((ext_vector_type(2))) float v2f;
typedef __attribute__((ext_vector_type(8))) float v8f;

constexpr int NB = 32;        // batch
constexpr int L  = 512;       // phoneme length
constexpr int H  = 512;       // hidden
constexpr int T  = 2048;      // max_len frames
constexpr int F  = 256;       // filter size
constexpr int CHUNK = 512;    // predictor chunk
constexpr int S  = NB * (T / CHUNK);   // 128 predictor samples
constexpr int KBLK = 32;
constexpr int LDSP = 36;      // padded LDS row stride (float4-aligned, bank-conflict-free)

// ------------------------------------------------------------------
// 1) Length regulator: cumsum scan + searchsorted gather
// ------------------------------------------------------------------
__global__ __launch_bounds__(512) void lr_scan_kernel(const float* __restrict__ dur,
                                                      int* __restrict__ cum) {
    __shared__ int sc[L];
    int b = blockIdx.x, l = threadIdx.x;
    float d = dur[b * L + l];
    sc[l] = (int)rintf(fmaxf(d, 0.0f));   // round-half-even == jnp.round
    __syncthreads();
    for (int off = 1; off < L; off <<= 1) {
        int v = (l >= off) ? sc[l - off] : 0;
        __syncthreads();
        sc[l] += v;
        __syncthreads();
    }
    cum[b * L + l] = sc[l];
}

__global__ __launch_bounds__(128) void lr_gather_kernel(const float* __restrict__ x,
                                                        const int* __restrict__ cum,
                                                        float* __restrict__ out) {
    int row = blockIdx.x;                  // b*T + t
    int b = row >> 11, t = row & (T - 1);
    const int* c = cum + b * L;
    int total = c[L - 1];
    // searchsorted(cum, t, side='right'): first j with cum[j] > t
    int lo = 0, hi = L;
    while (lo < hi) { int mid = (lo + hi) >> 1; if (c[mid] > t) hi = mid; else lo = mid + 1; }
    bool valid = t < total;
    int idx = min(lo, L - 1);
    const float4* src = (const float4*)(x + ((size_t)b * L + idx) * H);
    float4* dst = (float4*)(out + (size_t)row * H);
    float4 v = valid ? src[threadIdx.x] : make_float4(0.f, 0.f, 0.f, 0.f);
    dst[threadIdx.x] = v;
}

// ------------------------------------------------------------------
// 2) Conv1d(k=3, SAME) as fp32 WMMA GEMM:
//    O[l,f] = sum_tap sum_c X[l+tap-1,c]*W[f,c,tap] + bias[f]
//    X layout (S, CHUNK, CIN) row-major; O layout (S, CHUNK, F).
//    Block = 256 threads (8 waves), 128x64 output tile.
//    Each wave: 32x32 subtile = 2x2 accumulators (A and B fragment reuse).
//    Both As and Bs stored K-fastest so fragment K-pairs are adjacent dwords
//    (single ds_load_2addr per fragment, no repacking movs).
// ------------------------------------------------------------------
template <int CIN>
__global__ __launch_bounds__(256) void conv_gemm_kernel(const float* __restrict__ X,
                                                        const float* __restrict__ W,
                                                        const float* __restrict__ bias,
                                                        float* __restrict__ O) {
    __shared__ float As[128][LDSP];   // [row l][channel k]
    __shared__ float Bs[64][LDSP];    // [filter f][channel k]

    const int s  = blockIdx.z;
    const int l0 = blockIdx.y * 128;
    const int f0 = blockIdx.x * 64;

    const int tid  = threadIdx.x;
    const int wave = tid >> 5;
    const int lane = tid & 31;
    const int mn   = lane & 15;
    const int half = lane >> 4;
    const int mrow = (wave & 3) * 32;      // 4 waves along M, 32 rows each
    const int ncol = (wave >> 2) * 32;     // 2 waves along N, 32 cols each

    v8f acc00 = {};   // (m+0 , n+0 )
    v8f acc01 = {};   // (m+0 , n+16)
    v8f acc10 = {};   // (m+16, n+0 )
    v8f acc11 = {};   // (m+16, n+16)

    const int ar   = tid >> 1;             // A stage: row 0..127
    const int ac16 = (tid & 1) * 16;       // A stage: 16 channels (4 x float4)
    const int bf   = tid & 63;             // B stage: filter
    const int bc8  = (tid >> 6) * 8;       // B stage: 8 channels

    for (int tap = 0; tap < 3; ++tap) {
        const int lsrc = l0 + ar + tap - 1;                 // SAME padding, per chunk
        const bool okrow = (lsrc >= 0) && (lsrc < CHUNK);
        const float* xrow = X + ((size_t)s * CHUNK + (okrow ? lsrc : 0)) * CIN;
        for (int c0 = 0; c0 < CIN; c0 += KBLK) {
            // ---- stage A: 128 rows x KBLK channels (float4 stores, aligned) ----
            if (okrow) {
                #pragma unroll
                for (int j = 0; j < 4; ++j) {
                    float4 v = *(const float4*)(xrow + c0 + ac16 + 4 * j);
                    *(float4*)&As[ar][ac16 + 4 * j] = v;
                }
            } else {
                #pragma unroll
                for (int j = 0; j < 4; ++j)
                    *(float4*)&As[ar][ac16 + 4 * j] = make_float4(0.f, 0.f, 0.f, 0.f);
            }
            // ---- stage B: 64 filters x KBLK channels (weights, stride-3 over c) ----
            {
                float t[8];
                #pragma unroll
                for (int j = 0; j < 8; ++j)   // W layout (F, CIN, 3)
                    t[j] = W[((size_t)(f0 + bf) * CIN + (c0 + bc8 + j)) * 3 + tap];
                *(float4*)&Bs[bf][bc8 + 0] = make_float4(t[0], t[1], t[2], t[3]);
                *(float4*)&Bs[bf][bc8 + 4] = make_float4(t[4], t[5], t[6], t[7]);
            }
            __syncthreads();
            #pragma unroll
            for (int cc = 0; cc < KBLK; cc += 4) {
                const int k2 = cc + half * 2;   // lane's K pair: k2, k2+1
                v2f a0, a1, b0, b1;
                a0.x = As[mrow + mn][k2];        a0.y = As[mrow + mn][k2 + 1];
                a1.x = As[mrow + 16 + mn][k2];   a1.y = As[mrow + 16 + mn][k2 + 1];
                b0.x = Bs[ncol + mn][k2];        b0.y = Bs[ncol + mn][k2 + 1];
                b1.x = Bs[ncol + 16 + mn][k2];   b1.y = Bs[ncol + 16 + mn][k2 + 1];
                acc00 = __builtin_amdgcn_wmma_f32_16x16x4_f32(false, a0, false, b0,
                                                              (short)0, acc00, false, false);
                acc01 = __builtin_amdgcn_wmma_f32_16x16x4_f32(false, a0, false, b1,
                                                              (short)0, acc01, false, false);
                acc10 = __builtin_amdgcn_wmma_f32_16x16x4_f32(false, a1, false, b0,
                                                              (short)0, acc10, false, false);
                acc11 = __builtin_amdgcn_wmma_f32_16x16x4_f32(false, a1, false, b1,
                                                              (short)0, acc11, false, false);
            }
            __syncthreads();
        }
    }

    const int fa = f0 + ncol + mn;
    const int fb = fa + 16;
    const float ba = bias[fa];
    const float bb = bias[fb];
    #pragma unroll
    for (int r = 0; r < 8; ++r) {            // C/D layout: VGPR r -> M = r + 8*half
        int la = l0 + mrow + half * 8 + r;
        int lb = la + 16;
        float* rowa = O + ((size_t)s * CHUNK + la) * F;
        float* rowb = O + ((size_t)s * CHUNK + lb) * F;
        rowa[fa] = acc00[r] + ba;
        rowa[fb] = acc01[r] + bb;
        rowb[fa] = acc10[r] + ba;
        rowb[fb] = acc11[r] + bb;
    }
}

// ------------------------------------------------------------------
// 3) GroupNorm(num_groups=1) stats per sample over (C=256, L=512)
// ------------------------------------------------------------------
__global__ __launch_bounds__(256) void gn_stats_kernel(const float* __restrict__ h,
                                                       float* __restrict__ stats) {
    constexpr int N = CHUNK * F;  // 131072
    __shared__ float ss[256], sq[256];
    int s = blockIdx.x, tid = threadIdx.x;
    const float* p = h + (size_t)s * N;
    float a = 0.f, q = 0.f;
    for (int i = tid; i < N; i += 256) { float v = p[i]; a += v; q += v * v; }
    ss[tid] = a; sq[tid] = q;
    __syncthreads();
    for (int off = 128; off > 0; off >>= 1) {
        if (tid < off) { ss[tid] += ss[tid + off]; sq[tid] += sq[tid + off]; }
        __syncthreads();
    }
    if (tid == 0) {
        float mean = ss[0] / (float)N;
        float var  = sq[0] / (float)N - mean * mean;
        stats[2 * s]     = mean;
        stats[2 * s + 1] = rsqrtf(var + 1e-5f);
    }
}

// GN affine + ReLU, in place. h layout (S, CHUNK, F): channel = f (fastest dim).
__global__ __launch_bounds__(256) void gn_apply_kernel(float* __restrict__ h,
                                                       const float* __restrict__ stats,
                                                       const float* __restrict__ gw,
                                                       const float* __restrict__ gb) {
    int i4 = blockIdx.x * blockDim.x + threadIdx.x;  // float4 index
    int e  = i4 * 4;
    int s  = e >> 17;                                // / (CHUNK*F)
    int c  = e & (F - 1);
    float mean = stats[2 * s], rstd = stats[2 * s + 1];
    float4 v  = ((float4*)h)[i4];
    float4 w  = *(const float4*)(gw + c);
    float4 bb = *(const float4*)(gb + c);
    v.x = fmaxf((v.x - mean) * rstd * w.x + bb.x, 0.f);
    v.y = fmaxf((v.y - mean) * rstd * w.y + bb.y, 0.f);
    v.z = fmaxf((v.z - mean) * rstd * w.z + bb.z, 0.f);
    v.w = fmaxf((v.w - mean) * rstd * w.w + bb.w, 0.f);
    ((float4*)h)[i4] = v;
}

// ------------------------------------------------------------------
// 4) Fused GN2 + ReLU + Linear(256->1): pred[row] = sum_f relu(gn(h2))*lw[f] + lb
//    One wave32 per row; 8 waves per block.
// ------------------------------------------------------------------
__global__ __launch_bounds__(256) void rowdot_kernel(const float* __restrict__ h2,
                                                     const float* __restrict__ stats,
                                                     const float* __restrict__ gw,
                                                     const float* __restrict__ gb,
                                                     const float* __restrict__ lw,
                                                     const float* __restrict__ lb,
                                                     float* __restrict__ pred) {
    int wave = threadIdx.x >> 5, lane = threadIdx.x & 31;
    int row = blockIdx.x * 8 + wave;     // 0..S*CHUNK-1
    int s = row >> 9;                    // / CHUNK
    float mean = stats[2 * s], rstd = stats[2 * s + 1];
    const float* p = h2 + (size_t)row * F;
    float sum = 0.f;
    for (int f = lane; f < F; f += 32) {
        float v = fmaxf((p[f] - mean) * rstd * gw[f] + gb[f], 0.f);
        sum += v * lw[f];
    }
    #pragma unroll
    for (int off = 16; off > 0; off >>= 1) sum += __shfl_down(sum, off, 32);
    if (lane == 0) pred[row] = sum + lb[0];
}

// ------------------------------------------------------------------
// 5) Bucketize (searchsorted left into linspace(0,1,255)) + embedding add
// ------------------------------------------------------------------
__device__ __forceinline__ int bucketize255(float p) {
    int idx = 0;
    #pragma unroll 51
    for (int i = 0; i < 255; ++i) idx += ((float)i * (1.0f / 254.0f) < p) ? 1 : 0;
    return idx;   // in [0, 255]
}

__global__ __launch_bounds__(128) void embadd_kernel(float* __restrict__ out,
                                                     const float* __restrict__ pp,
                                                     const float* __restrict__ ep,
                                                     const float* __restrict__ pe,
                                                     const float* __restrict__ ee) {
    __shared__ int sidx[2];
    int row = blockIdx.x;                // b*T + t  == predictor row s*CHUNK + l
    if (threadIdx.x == 0) sidx[0] = bucketize255(pp[row]);
    if (threadIdx.x == 1) sidx[1] = bucketize255(ep[row]);
    __syncthreads();
    int pi = sidx[0], ei = sidx[1];
    float4* o = (float4*)(out + (size_t)row * H);
    const float4* a = (const float4*)(pe + (size_t)pi * H);
    const float4* b = (const float4*)(ee + (size_t)ei * H);
    float4 v = o[threadIdx.x], va = a[threadIdx.x], vb = b[threadIdx.x];
    v.x += va.x + vb.x;
    v.y += va.y + vb.y;
    v.z += va.z + vb.z;
    v.w += va.w + vb.w;
    o[threadIdx.x] = v;
}

// ------------------------------------------------------------------
// Host launcher
// ------------------------------------------------------------------
extern "C" void kernel_launch(void* const* d_in, const int* in_sizes, int n_in,
                              void* d_out, int out_size, void* d_ws, size_t ws_size,
                              hipStream_t stream) {
    (void)in_sizes; (void)n_in; (void)out_size; (void)ws_size;
    const float* x   = (const float*)d_in[0];   // (32,512,512)
    const float* dur = (const float*)d_in[1];   // (32,512)
    const float* Pp[10];
    const float* Pe[10];
    for (int i = 0; i < 10; ++i) {
        Pp[i] = (const float*)d_in[2 + i];      // pitch_pred params in dict order
        Pe[i] = (const float*)d_in[12 + i];     // energy_pred params
    }
    const float* pemb = (const float*)d_in[22]; // (256,512)
    const float* eemb = (const float*)d_in[23]; // (256,512)
    float* out = (float*)d_out;                 // (32,2048,512) == frames, updated in place

    float* ws    = (float*)d_ws;
    int*   cum   = (int*)ws;                    // 32*512 ints
    float* stats = ws + 16384;                  // 128 * {mean, rstd}
    float* predp = ws + 16384 + 256;            // 65536
    float* prede = predp + 65536;               // 65536
    float* h1    = prede + 65536;               // 128*512*256
    float* h2    = h1 + (size_t)S * CHUNK * F;  // 128*512*256

    // 1) length regulate -> frames in d_out
    lr_scan_kernel<<<NB, L, 0, stream>>>(dur, cum);
    lr_gather_kernel<<<NB * T, H / 4, 0, stream>>>(x, cum, out);

    // 2) predictors (sequential on stream; stats buffer reused safely)
    auto predictor = [&](const float* const* P, float* pred) {
        conv_gemm_kernel<H><<<dim3(F / 64, CHUNK / 128, S), 256, 0, stream>>>(out, P[0], P[1], h1);
        gn_stats_kernel<<<S, 256, 0, stream>>>(h1, stats);
        gn_apply_kernel<<<(S * CHUNK * F / 4) / 256, 256, 0, stream>>>(h1, stats, P[2], P[3]);
        conv_gemm_kernel<F><<<dim3(F / 64, CHUNK / 128, S), 256, 0, stream>>>(h1, P[4], P[5], h2);
        gn_stats_kernel<<<S, 256, 0, stream>>>(h2, stats);
        rowdot_kernel<<<(S * CHUNK) / 8, 256, 0, stream>>>(h2, stats, P[6], P[7], P[8], P[9], pred);
    };
    predictor(Pp, predp);
    predictor(Pe, prede);

    // 3) bucketize + embedding add into d_out
    embadd_kernel<<<NB * T, H / 4, 0, stream>>>(out, predp, prede, pemb, eemb);
}